// GATLayer_58944131170953
// MI455X (gfx1250) — compile-verified
//
#include <hip/hip_runtime.h>

typedef __attribute__((ext_vector_type(2))) float v2f;
typedef __attribute__((ext_vector_type(8))) float v8f;

#define IN_DIM 128
#define HD 128          // HEADS * OUT_DIM
#define HEADS 4
#define ODIM 32

// ---------------------------------------------------------------------------
// Zero-fill (harness poisons d_out / d_ws; we must re-init every call)
// ---------------------------------------------------------------------------
__global__ void zero_f32(float* __restrict__ p, size_t count) {
    size_t i = (size_t)blockIdx.x * blockDim.x + threadIdx.x;
    size_t stride = (size_t)gridDim.x * blockDim.x;
    for (; i < count; i += stride) p[i] = 0.0f;
}

// ---------------------------------------------------------------------------
// Kernel 1: h = x @ W^T via V_WMMA_F32_16X16X4_F32.
// One wave per 16x16 output tile; block = 256 threads = 8 waves = the 8
// column tiles of the 128-wide output for a 16-row strip of nodes.
// A(16x4) layout: lane L -> m = L&15, K base = 2*(L>>4), 2 floats.
// B(4x16) layout: lane L -> n = L&15, K base = 2*(L>>4), 2 floats (B[k][n] = W[n][k]).
// D(16x16): lane L -> n = L&15, row = r + 8*(L>>4) for VGPR r.
// ---------------------------------------------------------------------------
__global__ __launch_bounds__(256) void gemm_xWt_wmma(
        const float* __restrict__ x, const float* __restrict__ W,
        float* __restrict__ h) {
    const int lane = threadIdx.x & 31;
    const int wave = threadIdx.x >> 5;          // 0..7 : column tile
    const int rowBase = blockIdx.x * 16;
    const int colBase = wave * 16;
    const int m    = lane & 15;
    const int half = lane >> 4;                 // 0 or 1

    const float* __restrict__ xrow = x + (size_t)(rowBase + m) * IN_DIM;
    const float* __restrict__ wrow = W + (size_t)(colBase + m) * IN_DIM; // n == lane&15

    v8f acc = {};
    #pragma unroll 8
    for (int kk = 0; kk < IN_DIM; kk += 4) {
        const int k0 = kk + 2 * half;
        v2f a = *(const v2f*)(xrow + k0);   // A: x[rowBase+m][k0..k0+1]
        v2f b = *(const v2f*)(wrow + k0);   // B: W[colBase+n][k0..k0+1]
        acc = __builtin_amdgcn_wmma_f32_16x16x4_f32(
            /*neg_a=*/false, a, /*neg_b=*/false, b,
            /*c_mod=*/(short)0, acc, /*reuse_a=*/false, /*reuse_b=*/false);
    }

    float* __restrict__ ho = h + (size_t)(rowBase + 8 * half) * HD + colBase + m;
    #pragma unroll
    for (int r = 0; r < 8; ++r)
        ho[(size_t)r * HD] = acc[r];
}

// Scalar tail for rows not covered by full 16-row tiles (unused for N=100000).
__global__ void gemm_tail(const float* __restrict__ x, const float* __restrict__ W,
                          float* __restrict__ h, int rowStart, int n) {
    int idx = blockIdx.x * blockDim.x + threadIdx.x;
    int row = rowStart + idx / HD;
    int j   = idx % HD;
    if (row >= n) return;
    float acc = 0.f;
    const float* xr = x + (size_t)row * IN_DIM;
    const float* wr = W + (size_t)j * IN_DIM;
    for (int k = 0; k < IN_DIM; ++k) acc += xr[k] * wr[k];
    h[(size_t)row * HD + j] = acc;
}

// ---------------------------------------------------------------------------
// Kernel 2: per-node attention scores  s_src[n,h] = <h[n,h,:], a_src[h,:]>
// One thread per (node, head); float4 vector loads (32 floats per head).
// ---------------------------------------------------------------------------
__global__ void node_scores(const float* __restrict__ h,
                            const float* __restrict__ a_src,
                            const float* __restrict__ a_dst,
                            float* __restrict__ s_src,
                            float* __restrict__ s_dst, int n) {
    int idx = blockIdx.x * blockDim.x + threadIdx.x;  // node*HEADS + head
    if (idx >= n * HEADS) return;
    int head = idx & (HEADS - 1);
    int node = idx >> 2;
    const float4* hv = (const float4*)(h + (size_t)node * HD + head * ODIM);
    const float4* as = (const float4*)(a_src + head * ODIM);
    const float4* ad = (const float4*)(a_dst + head * ODIM);
    float ss = 0.f, sd = 0.f;
    #pragma unroll
    for (int i = 0; i < ODIM / 4; ++i) {
        float4 v = hv[i], s = as[i], d = ad[i];
        ss += v.x * s.x + v.y * s.y + v.z * s.z + v.w * s.w;
        sd += v.x * d.x + v.y * d.y + v.z * d.z + v.w * d.w;
    }
    s_src[idx] = ss;
    s_dst[idx] = sd;
}

__device__ __forceinline__ float leaky02(float v) {
    return v > 0.f ? v : 0.2f * v;
}

// ---------------------------------------------------------------------------
// Kernel 3: per-edge logits -> segment max over dst.
// e_max init = 0 and .at[dst].max includes init, so result >= 0 always:
// integer atomicMax on the raw bits is monotone for non-negative floats,
// and non-positive logits can never beat the 0 init, so we skip them.
// ---------------------------------------------------------------------------
__global__ void edge_max(const int* __restrict__ src, const int* __restrict__ dst,
                         const float* __restrict__ s_src, const float* __restrict__ s_dst,
                         float* __restrict__ e_max, int E) {
    int e = blockIdx.x * blockDim.x + threadIdx.x;
    if (e >= E) return;
    int s = src[e], d = dst[e];
    float4 a = *(const float4*)(s_src + (size_t)s * HEADS);
    float4 b = *(const float4*)(s_dst + (size_t)d * HEADS);
    float v[4] = { leaky02(a.x + b.x), leaky02(a.y + b.y),
                   leaky02(a.z + b.z), leaky02(a.w + b.w) };
    int* mp = (int*)(e_max + (size_t)d * HEADS);
    #pragma unroll
    for (int hh = 0; hh < HEADS; ++hh)
        if (v[hh] > 0.f) atomicMax(mp + hh, __float_as_int(v[hh]));
}

// ---------------------------------------------------------------------------
// Kernel 4: e_exp = exp(e - e_max[dst]); atomic segment-sum; cache numerators.
// ---------------------------------------------------------------------------
__global__ void edge_exp(const int* __restrict__ src, const int* __restrict__ dst,
                         const float* __restrict__ s_src, const float* __restrict__ s_dst,
                         const float* __restrict__ e_max, float* __restrict__ e_sum,
                         float* __restrict__ ex, int E) {
    int e = blockIdx.x * blockDim.x + threadIdx.x;
    if (e >= E) return;
    int s = src[e], d = dst[e];
    float4 a = *(const float4*)(s_src + (size_t)s * HEADS);
    float4 b = *(const float4*)(s_dst + (size_t)d * HEADS);
    float4 m = *(const float4*)(e_max + (size_t)d * HEADS);
    float4 r;
    r.x = __expf(leaky02(a.x + b.x) - m.x);
    r.y = __expf(leaky02(a.y + b.y) - m.y);
    r.z = __expf(leaky02(a.z + b.z) - m.z);
    r.w = __expf(leaky02(a.w + b.w) - m.w);
    *(float4*)(ex + (size_t)e * HEADS) = r;
    float* sp = e_sum + (size_t)d * HEADS;
    atomicAdd(sp + 0, r.x);
    atomicAdd(sp + 1, r.y);
    atomicAdd(sp + 2, r.z);
    atomicAdd(sp + 3, r.w);
}

// ---------------------------------------------------------------------------
// Kernel 5: weighted message aggregation. One wave32 per edge: lane l owns
// features [4l, 4l+4) -> one coalesced float4 gather of h[src] per lane and
// 4 fp32 atomic adds into out[dst]. h (51 MB) and out (51 MB) both live in
// the 192 MB L2, so this phase runs at L2 gather/atomic throughput.
// ---------------------------------------------------------------------------
__global__ __launch_bounds__(256) void edge_aggregate(
        const int* __restrict__ src, const int* __restrict__ dst,
        const float* __restrict__ h, const float* __restrict__ ex,
        const float* __restrict__ e_sum, float* __restrict__ out, int E) {
    int e = blockIdx.x * 8 + (threadIdx.x >> 5);
    if (e >= E) return;
    const int lane = threadIdx.x & 31;
    int s = src[e], d = dst[e];
    int head = lane >> 3;                       // 32 floats per head / 4 per lane
    float num = ex[(size_t)e * HEADS + head];
    float den = e_sum[(size_t)d * HEADS + head] + 1e-8f;
    float alpha = num / den;
    float4 v = *(const float4*)(h + (size_t)s * HD + lane * 4);
    float* o = out + (size_t)d * HD + lane * 4;
    atomicAdd(o + 0, v.x * alpha);
    atomicAdd(o + 1, v.y * alpha);
    atomicAdd(o + 2, v.z * alpha);
    atomicAdd(o + 3, v.w * alpha);
}

// ---------------------------------------------------------------------------
extern "C" void kernel_launch(void* const* d_in, const int* in_sizes, int n_in,
                              void* d_out, int out_size, void* d_ws, size_t ws_size,
                              hipStream_t stream) {
    const float* x     = (const float*)d_in[0];
    const int*   edges = (const int*)  d_in[1];
    const float* W     = (const float*)d_in[2];
    const float* a_src = (const float*)d_in[3];
    const float* a_dst = (const float*)d_in[4];

    const int n = in_sizes[0] / IN_DIM;
    const int E = in_sizes[1] / 2;
    const int* src = edges;
    const int* dst = edges + E;

    float* out = (float*)d_out;

    // Workspace carving (all regions 16B-aligned: counts are multiples of 4).
    float* ws    = (float*)d_ws;
    float* h     = ws;                               // n * 128
    float* s_src = h     + (size_t)n * HD;           // n * 4
    float* s_dst = s_src + (size_t)n * HEADS;        // n * 4
    float* e_max = s_dst + (size_t)n * HEADS;        // n * 4
    float* e_sum = e_max + (size_t)n * HEADS;        // n * 4  (contiguous w/ e_max)
    float* ex    = e_sum + (size_t)n * HEADS;        // E * 4

    // Re-init accumulators every call (harness poisons buffers).
    zero_f32<<<2048, 256, 0, stream>>>(out, (size_t)n * HD);
    zero_f32<<<512,  256, 0, stream>>>(e_max, (size_t)n * HEADS * 2); // e_max + e_sum

    // 1) h = x @ W^T (WMMA f32 16x16x4)
    const int nTiles = n / 16;
    if (nTiles > 0)
        gemm_xWt_wmma<<<nTiles, 256, 0, stream>>>(x, W, h);
    const int rem = n - nTiles * 16;
    if (rem > 0)
        gemm_tail<<<(rem * HD + 255) / 256, 256, 0, stream>>>(x, W, h, nTiles * 16, n);

    // 2) per-node attention scores
    node_scores<<<(n * HEADS + 255) / 256, 256, 0, stream>>>(h, a_src, a_dst,
                                                             s_src, s_dst, n);
    // 3) segment max
    edge_max<<<(E + 255) / 256, 256, 0, stream>>>(src, dst, s_src, s_dst, e_max, E);
    // 4) exp + segment sum
    edge_exp<<<(E + 255) / 256, 256, 0, stream>>>(src, dst, s_src, s_dst,
                                                  e_max, e_sum, ex, E);
    // 5) aggregate
    edge_aggregate<<<(E + 7) / 8, 256, 0, stream>>>(src, dst, h, ex, e_sum, out, E);
}